// PoolAttn_38646115730123
// MI455X (gfx1250) — compile-verified
//
#include <hip/hip_runtime.h>

typedef __attribute__((ext_vector_type(2))) float v2f;
typedef __attribute__((ext_vector_type(4))) float v4f;
typedef __attribute__((ext_vector_type(8))) float v8f;

#define NB   16
#define NC   256
#define NH   64
#define NW   64
#define NHW  4096          // 64*64
#define NCHW (256*4096)

// ---------------------------------------------------------------------------
// K1: pooled projections for the patch branch.
// p1[b,c,h,q] = mean_{w in [16q,16q+16)} x[b,c,h,w]      -> [B,C,64,4]
// p2[b,c,k,w] = mean_{h in [16k,16k+16)} x[b,c,h,w]      -> [B,C,4,64]
// ---------------------------------------------------------------------------
__global__ __launch_bounds__(256) void pool_kernel(const float* __restrict__ x,
                                                   float* __restrict__ p1,
                                                   float* __restrict__ p2) {
    const int bc = blockIdx.x;            // 0 .. B*C-1
    const int t  = threadIdx.x;           // 0 .. 255
    const float* xp = x + (size_t)bc * NHW;

    // role A: one (h, quarter) per thread -- 64B-aligned float4 stream
    {
        const int h = t >> 2, q = t & 3;
        const v4f* r = (const v4f*)(xp + h * NW + q * 16);
        float s = 0.f;
        #pragma unroll
        for (int i = 0; i < 4; ++i) {
            const v4f v = r[i];
            s += v.x + v.y + v.z + v.w;
        }
        p1[(size_t)bc * 256 + h * 4 + q] = s * 0.0625f;
    }
    // role B: one (quarter, w) per thread (coalesced across lanes per step)
    {
        const int k = t >> 6, w = t & 63;
        const float* col = xp + (k * 16) * NW + w;
        float s = 0.f;
        #pragma unroll
        for (int i = 0; i < 16; ++i) s += col[i * NW];
        p2[(size_t)bc * 256 + k * NW + w] = s * 0.0625f;
    }
}

// ---------------------------------------------------------------------------
// K2: embedding-dim branch (before conv1). One thread per pixel; channel
// vector is read with stride HW (coalesced across lanes). Two passes over x;
// second pass hits L2 (whole x = 64 MiB < 192 MB global L2).
// ---------------------------------------------------------------------------
__global__ __launch_bounds__(256) void emb_kernel(const float* __restrict__ x,
                                                  float* __restrict__ e) {
    const int pix = blockIdx.x * 256 + threadIdx.x;   // 0 .. B*HW-1
    const int b   = pix >> 12;
    const int hw  = pix & (NHW - 1);
    const float* xp = x + (size_t)b * NCHW + hw;
    float* ep       = e + (size_t)b * NCHW + hw;

    float e2[4][8];
    #pragma unroll
    for (int k = 0; k < 4; ++k) {
        float acc[8];
        #pragma unroll
        for (int d = 0; d < 8; ++d) acc[d] = 0.f;
        for (int ii = 0; ii < 8; ++ii) {
            const float* q = xp + (size_t)((k * 8 + ii) * 8) * NHW;
            #pragma unroll
            for (int d = 0; d < 8; ++d) acc[d] += q[(size_t)d * NHW];
        }
        #pragma unroll
        for (int d = 0; d < 8; ++d) e2[k][d] = acc[d] * 0.125f;
    }

    for (int i = 0; i < 32; ++i) {
        const float* q = xp + (size_t)(i * 8) * NHW;
        float xr[8];
        #pragma unroll
        for (int d = 0; d < 8; ++d) xr[d] = q[(size_t)d * NHW];
        const float e10 = 0.5f * (xr[0] + xr[1]);
        const float e11 = 0.5f * (xr[2] + xr[3]);
        const float e12 = 0.5f * (xr[4] + xr[5]);
        const float e13 = 0.5f * (xr[6] + xr[7]);
        float* o = ep + (size_t)(i * 8) * NHW;
        #pragma unroll
        for (int d = 0; d < 8; ++d)
            o[(size_t)d * NHW] = e10 * e2[0][d] + e11 * e2[1][d] +
                                 e12 * e2[2][d] + e13 * e2[3][d];
    }
}

// ---------------------------------------------------------------------------
// K3: patch-branch attention via V_WMMA_F32_16X16X4_F32 + fused conv0.
// 16 waves per (b,c); each wave = one 16x16 tile, one WMMA (K=4).
// LDS tile is zero-padded (66 rows x 68 stride, rounded up to 9*512 so the
// zero pass is 9 unconditional stores per thread); WMMA results land at
// (+1,+1), so the 3x3 depthwise conv0 is a branch-free 9-tap FMA chain.
// ---------------------------------------------------------------------------
#define STR3 68
__global__ __launch_bounds__(512) void patch_wmma_conv0(
        const float* __restrict__ p1, const float* __restrict__ p2,
        const float* __restrict__ w0, const float* __restrict__ b0,
        float* __restrict__ acc) {
    __shared__ float smem[9 * 512];   // >= 66*68 = 4488

    const int bc   = blockIdx.x;
    const int c    = bc & (NC - 1);
    const int tid  = threadIdx.x;
    const int lane = tid & 31;
    const int wave = tid >> 5;        // 0..15
    const int tm   = wave >> 2;       // tile row 0..3
    const int tn   = wave & 3;        // tile col 0..3

    // A fragment (rows tm*16 .. +15 of [64,4]): lanes 0-15 K0/K1, 16-31 K2/K3
    const int m  = tm * 16 + (lane & 15);
    const int kb = (lane >> 4) * 2;   // 0 or 2
    const v2f a = *(const v2f*)(p1 + (size_t)bc * 256 + m * 4 + kb);

    // B fragment (cols tn*16 .. +15 of [4,64])
    const int n = lane & 15;
    const float* p2c = p2 + (size_t)bc * 256;
    v2f bf;
    bf.x = p2c[(kb + 0) * NW + tn * 16 + n];
    bf.y = p2c[(kb + 1) * NW + tn * 16 + n];

    v8f cz = {};
    v8f d = __builtin_amdgcn_wmma_f32_16x16x4_f32(
        /*neg_a=*/false, a, /*neg_b=*/false, bf,
        /*c_mod=*/(short)0, cz, /*reuse_a=*/false, /*reuse_b=*/false);

    // branch-free zero of the padded tile
    #pragma unroll
    for (int i = 0; i < 9; ++i) smem[tid + i * 512] = 0.f;
    __syncthreads();

    const int rh = (lane >> 4) * 8;
    #pragma unroll
    for (int r = 0; r < 8; ++r)
        smem[(tm * 16 + rh + r + 1) * STR3 + (tn * 16 + n + 1)] = d[r];
    __syncthreads();

    float wv[9];
    #pragma unroll
    for (int i = 0; i < 9; ++i) wv[i] = w0[c * 9 + i];
    const float bias = b0[c];

    float* out = acc + (size_t)bc * NHW;
    #pragma unroll
    for (int i = 0; i < 8; ++i) {
        const int pix = tid + i * 512;
        const int h = pix >> 6, w = pix & 63;
        const float* t0 = &smem[h * STR3 + w];       // (h-1+dy, w-1+dx) -> +dy,+dx
        float s = bias;
        #pragma unroll
        for (int dy = 0; dy < 3; ++dy)
            #pragma unroll
            for (int dx = 0; dx < 3; ++dx)
                s += wv[dy * 3 + dx] * t0[dy * STR3 + dx];
        out[pix] = s;
    }
}

// ---------------------------------------------------------------------------
// K4a: zero GroupNorm accumulators (d_ws is poisoned by the harness).
// ---------------------------------------------------------------------------
__global__ void zero_stats(float* __restrict__ stats) {
    if (threadIdx.x < 32) stats[threadIdx.x] = 0.f;
}

// ---------------------------------------------------------------------------
// K4: acc += conv1(x_emb) + b1, fused GroupNorm sum/sum^2.
// LDS staging of the 6 input rows is done with GLOBAL_LOAD_ASYNC_TO_LDS_B32
// (memory -> LDS DMA path, ASYNCcnt-tracked, no VGPR round trip); halo rows
// outside the image are zeroed with plain ds stores. Reduction is an in-wave
// shuffle butterfly + one cross-wave pass + two float atomics per block (sum, sum^2).
// ---------------------------------------------------------------------------
#define STR6 66
__global__ __launch_bounds__(256) void conv1_add_stats(
        const float* __restrict__ e, const float* __restrict__ w1,
        const float* __restrict__ b1, float* __restrict__ acc,
        float* __restrict__ stats) {
    __shared__ float tile[6 * STR6];
    __shared__ float wsum1[8];
    __shared__ float wsum2[8];

    const int tid  = threadIdx.x;
    const int bc   = blockIdx.x >> 4;         // (b,c)
    const int h0   = (blockIdx.x & 15) * 4;   // first output row
    const int b    = bc >> 8;
    const int c    = bc & (NC - 1);
    const float* ch = e + (size_t)bc * NHW;
    const unsigned lds_base = (unsigned)(uintptr_t)&tile[0];   // LDS byte addr

    // stage rows h0-1 .. h0+4 (zero-padded top/bottom), cols at +1
    #pragma unroll
    for (int i = 0; i < 2; ++i) {
        const int t = tid + i * 256;
        if (t < 384) {
            const int r  = t >> 6;            // 0..5
            const int cl = t & 63;
            const int gh = h0 - 1 + r;
            const unsigned dst = lds_base + (unsigned)(r * STR6 + cl + 1) * 4u;
            if (gh >= 0 && gh < NH) {
                const int voff = (gh * NW + cl) * 4;   // byte offset from ch
                asm volatile("global_load_async_to_lds_b32 %0, %1, %2"
                             :: "v"(dst), "v"(voff), "s"(ch) : "memory");
            } else {
                tile[r * STR6 + cl + 1] = 0.f;
            }
        }
    }
    if (tid < 12) tile[(tid >> 1) * STR6 + (tid & 1) * 65] = 0.f;  // side pads
    asm volatile("s_wait_asynccnt 0x0" ::: "memory");
    __syncthreads();

    float wv[9];
    #pragma unroll
    for (int i = 0; i < 9; ++i) wv[i] = w1[c * 9 + i];

    const int hl = tid >> 6;                  // 0..3
    const int w  = tid & 63;
    const float* t0 = &tile[hl * STR6 + w];
    float s = b1[c];
    #pragma unroll
    for (int dy = 0; dy < 3; ++dy)
        #pragma unroll
        for (int dx = 0; dx < 3; ++dx)
            s += wv[dy * 3 + dx] * t0[dy * STR6 + dx];

    const size_t idx = (size_t)bc * NHW + (h0 + hl) * NW + w;
    const float v = acc[idx] + s;
    acc[idx] = v;

    // wave32 shuffle reduction for sum and sum-of-squares
    float s1 = v, s2 = v * v;
    #pragma unroll
    for (int off = 16; off > 0; off >>= 1) {
        s1 += __shfl_down(s1, off, 32);
        s2 += __shfl_down(s2, off, 32);
    }
    if ((tid & 31) == 0) { wsum1[tid >> 5] = s1; wsum2[tid >> 5] = s2; }
    __syncthreads();
    if (tid == 0) {
        float a1 = 0.f, a2 = 0.f;
        #pragma unroll
        for (int i = 0; i < 8; ++i) { a1 += wsum1[i]; a2 += wsum2[i]; }
        atomicAdd(&stats[2 * b + 0], a1);
        atomicAdd(&stats[2 * b + 1], a2);
    }
}

// ---------------------------------------------------------------------------
// K5: final conv2 over the GroupNorm output. The affine (y = acc*sc + tc)
// is applied WHILE staging into the zero-padded LDS tile, so the padding is
// exactly the reference's zero-padded y and the conv body is branch-free.
// ---------------------------------------------------------------------------
__global__ __launch_bounds__(256) void conv2_norm(
        const float* __restrict__ acc, const float* __restrict__ w2,
        const float* __restrict__ b2, const float* __restrict__ nw,
        const float* __restrict__ nb, const float* __restrict__ stats,
        float* __restrict__ out) {
    __shared__ float tile[6 * STR6];

    const int tid  = threadIdx.x;
    const int bc   = blockIdx.x >> 4;
    const int h0   = (blockIdx.x & 15) * 4;
    const int b    = bc >> 8;
    const int c    = bc & (NC - 1);
    const float* ch = acc + (size_t)bc * NHW;

    const float inv_n = 1.f / (float)NCHW;
    const float mean = stats[2 * b + 0] * inv_n;
    const float var  = stats[2 * b + 1] * inv_n - mean * mean;
    const float r    = rsqrtf(var + 1e-5f);
    const float sc   = nw[c] * r;
    const float tc   = nb[c] - mean * sc;

    #pragma unroll
    for (int i = 0; i < 2; ++i) {
        const int t = tid + i * 256;
        if (t < 384) {
            const int rr = t >> 6;
            const int cl = t & 63;
            const int gh = h0 - 1 + rr;
            float v = 0.f;
            if (gh >= 0 && gh < NH) v = fmaf(ch[gh * NW + cl], sc, tc);
            tile[rr * STR6 + cl + 1] = v;
        }
    }
    if (tid < 12) tile[(tid >> 1) * STR6 + (tid & 1) * 65] = 0.f;
    __syncthreads();

    float wv[9];
    #pragma unroll
    for (int i = 0; i < 9; ++i) wv[i] = w2[c * 9 + i];

    const int hl = tid >> 6;
    const int w  = tid & 63;
    const float* t0 = &tile[hl * STR6 + w];
    float s = b2[c];
    #pragma unroll
    for (int dy = 0; dy < 3; ++dy)
        #pragma unroll
        for (int dx = 0; dx < 3; ++dx)
            s += wv[dy * 3 + dx] * t0[dy * STR6 + dx];

    out[(size_t)bc * NHW + (h0 + hl) * NW + w] = s;
}

// ---------------------------------------------------------------------------
extern "C" void kernel_launch(void* const* d_in, const int* in_sizes, int n_in,
                              void* d_out, int out_size, void* d_ws, size_t ws_size,
                              hipStream_t stream) {
    (void)in_sizes; (void)n_in; (void)out_size; (void)ws_size;
    const float* x  = (const float*)d_in[0];
    const float* w0 = (const float*)d_in[1];
    const float* b0 = (const float*)d_in[2];
    const float* w1 = (const float*)d_in[3];
    const float* b1 = (const float*)d_in[4];
    const float* w2 = (const float*)d_in[5];
    const float* b2 = (const float*)d_in[6];
    const float* nw = (const float*)d_in[7];
    const float* nb = (const float*)d_in[8];
    float* out = (float*)d_out;

    char* ws = (char*)d_ws;
    float* p1    = (float*)(ws);                                // 4 MiB
    float* p2    = (float*)(ws + ((size_t)4 << 20));            // 4 MiB
    float* stats = (float*)(ws + ((size_t)8 << 20));            // 128 B
    float* acc   = (float*)(ws + ((size_t)8 << 20) + 4096);     // 64 MiB

    pool_kernel     <<<NB * NC, 256, 0, stream>>>(x, p1, p2);
    emb_kernel      <<<(NB * NHW) / 256, 256, 0, stream>>>(x, out);   // out = x_emb scratch
    patch_wmma_conv0<<<NB * NC, 512, 0, stream>>>(p1, p2, w0, b0, acc);
    zero_stats      <<<1, 32, 0, stream>>>(stats);
    conv1_add_stats <<<NB * NC * (NH / 4), 256, 0, stream>>>(out, w1, b1, acc, stats);
    conv2_norm      <<<NB * NC * (NH / 4), 256, 0, stream>>>(acc, w2, b2, nw, nb, stats, out);
}